// AttFusion_68882685493846
// MI455X (gfx1250) — compile-verified
//
#include <hip/hip_runtime.h>
#include <math.h>

#ifndef __has_builtin
#define __has_builtin(x) 0
#endif

// ---- CDNA5 async global->LDS path (ASYNCcnt), with safe fallback ----------
#if __has_builtin(__builtin_amdgcn_global_load_async_to_lds_b128)
#define ASYNC_LDS 1
#else
#define ASYNC_LDS 0
#endif

#if __has_builtin(__builtin_amdgcn_s_wait_asynccnt)
#define WAIT_ASYNC() __builtin_amdgcn_s_wait_asynccnt(0)
#elif ASYNC_LDS
#define WAIT_ASYNC() asm volatile("s_wait_asynccnt 0x0" ::: "memory")
#else
#define WAIT_ASYNC() ((void)0)
#endif

typedef int v4i __attribute__((vector_size(16)));
typedef __attribute__((address_space(1))) v4i* g_v4i_ptr;
typedef __attribute__((address_space(3))) v4i* l_v4i_ptr;

// Problem constants (from reference)
#define CDIM   256
#define HDIM   96
#define WDIM   288
#define PDIM   (HDIM * WDIM)      // 27648
#define BREC   4
#define LMAXD  6
#define TPX    16                 // pixels per tile (2 blocks / WGP)
#define ROWP   17                 // padded row stride (dwords) -> no LDS bank conflicts
#define NTILES (PDIM / TPX)       // 1728
#define THR    256
#define CGRP   (THR / TPX)        // 16 channel groups
#define CPG    (CDIM / CGRP)      // 16 channels per group

// LDS floats: padded xf tile + per-group partial scores + attention weights
#define LDS_XF   (LMAXD * CDIM * ROWP)       // 26112
#define LDS_SP   (CGRP * LMAXD * TPX)        // 1536
#define LDS_ATT  (LMAXD * TPX)               // 96
#define LDS_FLOATS (LDS_XF + LDS_SP + LDS_ATT)   // 27744 -> 110,976 B

__device__ __forceinline__ void stage16(const float* __restrict__ gsrc,
                                        float* __restrict__ ldst) {
#if ASYNC_LDS
  __builtin_amdgcn_global_load_async_to_lds_b128(
      (g_v4i_ptr)gsrc, (l_v4i_ptr)ldst, /*offset=*/0, /*cpol=*/0);
#else
  *(float4*)ldst = *(const float4*)gsrc;
#endif
}

__global__ __launch_bounds__(THR)
void att_fusion_kernel(const float* __restrict__ x,
                       const int*   __restrict__ record_len,
                       float*       __restrict__ out) {
  extern __shared__ float smem[];
  float* xfs = smem;                    // [LMAXD][CDIM][ROWP] (ROWP-padded rows)
  float* sp  = xfs + LDS_XF;            // [CGRP][LMAXD][TPX]
  float* att = sp + LDS_SP;             // [LMAXD][TPX]

  const int tid = threadIdx.x;
  const int blk = blockIdx.x;
  const int b   = blk / NTILES;
  const int p0  = (blk - b * NTILES) * TPX;

  // Uniform: record length, offset (from device record_len)
  int len = 1, off = 0;
#pragma unroll
  for (int i = 0; i < BREC; ++i) {
    const int r = record_len[i];
    if (i < b) off += r;
    if (i == b) len = r;
  }

  // ---- Stage xf tile via async global->LDS: one HBM read of x -------------
#pragma unroll
  for (int l = 0; l < LMAXD; ++l) {
    const int lc = (l < len - 1) ? l : (len - 1);    // index clamp
    const size_t gbase = (size_t)(off + lc) * (size_t)CDIM * (size_t)PDIM;
#pragma unroll
    for (int k = tid; k < (CDIM * TPX / 4); k += THR) {   // 1024 float4 / l
      const int c = k >> 2;          // TPX/4 == 4 float4 per channel row
      const int j = k & 3;
      const float* gsrc = x + gbase + (size_t)c * PDIM + p0 + j * 4;
      float* ldst = xfs + (l * CDIM + c) * ROWP + j * 4;
      stage16(gsrc, ldst);
    }
  }
  WAIT_ASYNC();
  __syncthreads();

  const int tp = tid & (TPX - 1);
  const int g  = tid >> 4;             // channel group

  // ---- Pass 1: partial scores over this thread's channel slice ------------
  float s[LMAXD];
#pragma unroll
  for (int l = 0; l < LMAXD; ++l) s[l] = 0.0f;

  for (int cc = 0; cc < CPG; ++cc) {
    const int c = g * CPG + cc;
    const float* col = xfs + c * ROWP + tp;   // l stride = CDIM*ROWP
    const float q = col[0];                   // l == 0 row is q
#pragma unroll
    for (int l = 0; l < LMAXD; ++l)
      s[l] = fmaf(q, col[l * CDIM * ROWP], s[l]);
  }
#pragma unroll
  for (int l = 0; l < LMAXD; ++l)
    sp[(g * LMAXD + l) * TPX + tp] = s[l];
  __syncthreads();

  // ---- Fixed-order reduction + masked softmax (deterministic) -------------
  if (tid < TPX) {
    float t[LMAXD];
#pragma unroll
    for (int l = 0; l < LMAXD; ++l) {
      float acc = 0.0f;
#pragma unroll
      for (int g2 = 0; g2 < CGRP; ++g2)
        acc += sp[(g2 * LMAXD + l) * TPX + tid];
      t[l] = (l < len) ? acc * 0.0625f : -1.0e9f;   // 1/sqrt(256) = 0.0625
    }
    float m = t[0];
#pragma unroll
    for (int l = 1; l < LMAXD; ++l) m = fmaxf(m, t[l]);
    float sum = 0.0f;
#pragma unroll
    for (int l = 0; l < LMAXD; ++l) { t[l] = expf(t[l] - m); sum += t[l]; }
    const float inv = 1.0f / sum;
#pragma unroll
    for (int l = 0; l < LMAXD; ++l) att[l * TPX + tid] = t[l] * inv;
  }
  __syncthreads();

  // ---- Pass 2: weighted sum from LDS, coalesced store ---------------------
  float a[LMAXD];
#pragma unroll
  for (int l = 0; l < LMAXD; ++l) a[l] = att[l * TPX + tp];

  float* obase = out + (size_t)b * CDIM * PDIM + p0 + tp;
  for (int cc = 0; cc < CPG; ++cc) {
    const int c = g * CPG + cc;
    const float* col = xfs + c * ROWP + tp;
    float o = 0.0f;
#pragma unroll
    for (int l = 0; l < LMAXD; ++l)
      o = fmaf(a[l], col[l * CDIM * ROWP], o);
    obase[(size_t)c * PDIM] = o;
  }
}

extern "C" void kernel_launch(void* const* d_in, const int* in_sizes, int n_in,
                              void* d_out, int out_size, void* d_ws, size_t ws_size,
                              hipStream_t stream) {
  (void)in_sizes; (void)n_in; (void)out_size; (void)d_ws; (void)ws_size;
  const float* x  = (const float*)d_in[0];
  const int*   rl = (const int*)d_in[1];
  float*       o  = (float*)d_out;

  const size_t shmem = (size_t)LDS_FLOATS * sizeof(float);  // 110,976 B
  (void)hipFuncSetAttribute((const void*)att_fusion_kernel,
                            hipFuncAttributeMaxDynamicSharedMemorySize,
                            (int)shmem);

  dim3 grid(BREC * NTILES);   // 6912 blocks, 2 co-resident per WGP
  att_fusion_kernel<<<grid, THR, shmem, stream>>>(x, rl, o);
}